// ConduitNetwork_15341623181950
// MI455X (gfx1250) — compile-verified
//
#include <hip/hip_runtime.h>
#include <hip/hip_bf16.h>

// Physical constants (fp32, matching the reference's float32 math)
#define K_GRAVITY      9.81f
#define K_ICE_DENSITY  917.0f
#define K_STEP_HEIGHT  0.1f
#define K_MELT_CONST   (1.0f / (917.0f * 335000.0f))   // 1/(rho_i*L)
#define K_CLOSURE      (2.0f * 6e-24f / 27.0f)         // 2*A*n^-n

// clang ext_vector types: accepted by __builtin_nontemporal_load/store and
// lower to b128 global loads/stores with NT temporal hints.
typedef float __attribute__((ext_vector_type(4))) fx4;
typedef int   __attribute__((ext_vector_type(4))) ix4;

// Per-node AoS record: {overburden, sum, cnt, netflux} = 16B. The link-pass
// gather + 3 atomics per endpoint all hit ONE 64B cacheline. 4M records =
// 64 MB -> resident in the 192 MB L2. One-touch streams use NT hints so they
// never evict the records.

// ---------------------------------------------------------------------------
// Kernel A (vec4): records[i] = { rho_i*g*h - p_w, 0, 0, 0 }, 4 nodes/thread
// ---------------------------------------------------------------------------
__global__ void __launch_bounds__(256)
prep_v4(const fx4* __restrict__ ice_h,
        const fx4* __restrict__ p_w,
        fx4* __restrict__ rec,
        int nvec)
{
    int j = blockIdx.x * blockDim.x + threadIdx.x;
    if (j >= nvec) return;
    fx4 h  = __builtin_nontemporal_load(ice_h + j);
    fx4 pw = __builtin_nontemporal_load(p_w + j);
    const float c = K_ICE_DENSITY * K_GRAVITY;
    int b = j * 4;
    rec[b + 0] = (fx4){fmaf(c, h.x, -pw.x), 0.f, 0.f, 0.f};
    rec[b + 1] = (fx4){fmaf(c, h.y, -pw.y), 0.f, 0.f, 0.f};
    rec[b + 2] = (fx4){fmaf(c, h.z, -pw.z), 0.f, 0.f, 0.f};
    rec[b + 3] = (fx4){fmaf(c, h.w, -pw.w), 0.f, 0.f, 0.f};
}

__global__ void __launch_bounds__(64)
prep_tail(const float* __restrict__ ice_h,
          const float* __restrict__ p_w,
          fx4* __restrict__ rec,
          int start, int n)
{
    int i = start + blockIdx.x * blockDim.x + threadIdx.x;
    if (i >= n) return;
    float ob = fmaf(K_ICE_DENSITY * K_GRAVITY, ice_h[i], -p_w[i]);
    rec[i] = (fx4){ob, 0.f, 0.f, 0.f};
}

// ---------------------------------------------------------------------------
// Kernel B (vec4): 4 links per thread. b128 NT loads for all six streams,
// then per link: 2 L2-hot gathers + 6 same-cacheline atomics.
// ---------------------------------------------------------------------------
__device__ __forceinline__ void link_body(float* __restrict__ recf,
                                          int hn, int tn,
                                          float q, float g, float v, float a)
{
    size_t hb = (size_t)hn * 4u;
    size_t tb = (size_t)tn * 4u;
    float obh = recf[hb];
    float obt = recf[tb];
    float eff  = 0.5f * (obh + obt);
    float eff3 = eff * eff * eff;
    float rhs = fmaf(K_MELT_CONST * q, g,
                fmaf(v, K_STEP_HEIGHT, -(K_CLOSURE * eff3 * a)));
    atomicAdd(recf + hb + 1, rhs);
    atomicAdd(recf + hb + 2, 1.0f);
    atomicAdd(recf + hb + 3, q);
    atomicAdd(recf + tb + 1, rhs);
    atomicAdd(recf + tb + 2, 1.0f);
    atomicAdd(recf + tb + 3, -q);
}

__global__ void __launch_bounds__(256)
link_v4(const fx4* __restrict__ vel,
        const fx4* __restrict__ area,
        const fx4* __restrict__ grad,
        const fx4* __restrict__ flux,
        const ix4* __restrict__ head,
        const ix4* __restrict__ tail,
        float* __restrict__ recf,
        int nvec)
{
    int j = blockIdx.x * blockDim.x + threadIdx.x;
    if (j >= nvec) return;

    ix4 h4 = __builtin_nontemporal_load(head + j);
    ix4 t4 = __builtin_nontemporal_load(tail + j);
    fx4 q4 = __builtin_nontemporal_load(flux + j);
    fx4 g4 = __builtin_nontemporal_load(grad + j);
    fx4 v4 = __builtin_nontemporal_load(vel  + j);
    fx4 a4 = __builtin_nontemporal_load(area + j);

    link_body(recf, h4.x, t4.x, q4.x, g4.x, v4.x, a4.x);
    link_body(recf, h4.y, t4.y, q4.y, g4.y, v4.y, a4.y);
    link_body(recf, h4.z, t4.z, q4.z, g4.z, v4.z, a4.z);
    link_body(recf, h4.w, t4.w, q4.w, g4.w, v4.w, a4.w);
}

__global__ void __launch_bounds__(64)
link_tail(const float* __restrict__ vel,
          const float* __restrict__ area,
          const float* __restrict__ grad,
          const float* __restrict__ flux,
          const int*   __restrict__ head,
          const int*   __restrict__ tail,
          float* __restrict__ recf,
          int start, int nlinks)
{
    int i = start + blockIdx.x * blockDim.x + threadIdx.x;
    if (i >= nlinks) return;
    link_body(recf, head[i], tail[i], flux[i], grad[i], vel[i], area[i]);
}

// ---------------------------------------------------------------------------
// Kernel C (vec4): out = sum/max(cnt,1) + netflux - melt, 4 nodes/thread
// ---------------------------------------------------------------------------
__global__ void __launch_bounds__(256)
finalize_v4(const fx4* __restrict__ rec,
            const fx4* __restrict__ melt,
            fx4* __restrict__ out,
            int nvec)
{
    int j = blockIdx.x * blockDim.x + threadIdx.x;
    if (j >= nvec) return;
    int b = j * 4;
    fx4 r0 = rec[b + 0];
    fx4 r1 = rec[b + 1];
    fx4 r2 = rec[b + 2];
    fx4 r3 = rec[b + 3];
    fx4 m  = __builtin_nontemporal_load(melt + j);
    fx4 o;
    o.x = r0.y / fmaxf(r0.z, 1.0f) + r0.w - m.x;
    o.y = r1.y / fmaxf(r1.z, 1.0f) + r1.w - m.y;
    o.z = r2.y / fmaxf(r2.z, 1.0f) + r2.w - m.z;
    o.w = r3.y / fmaxf(r3.z, 1.0f) + r3.w - m.w;
    __builtin_nontemporal_store(o, out + j);
}

__global__ void __launch_bounds__(64)
finalize_tail(const fx4* __restrict__ rec,
              const float* __restrict__ melt,
              float* __restrict__ out,
              int start, int n)
{
    int i = start + blockIdx.x * blockDim.x + threadIdx.x;
    if (i >= n) return;
    fx4 r = rec[i];
    float o = r.y / fmaxf(r.z, 1.0f) + r.w - melt[i];
    __builtin_nontemporal_store(o, out + i);
}

// ---------------------------------------------------------------------------
// Launch wrapper. Input order (setup_inputs):
//  0 ice_thickness[N]  1 water_pressure[N]  2 meltwater_input[N]
//  3 ice_sliding_velocity[L]  4 conduit_area[L]  5 hydraulic_gradient[L]
//  6 water_flux[L]  7 node_at_link_head[L]  8 node_at_link_tail[L]
// Workspace: N fx4 records = 16N bytes (64 MB for N=4M).
// ---------------------------------------------------------------------------
extern "C" void kernel_launch(void* const* d_in, const int* in_sizes, int n_in,
                              void* d_out, int out_size, void* d_ws, size_t ws_size,
                              hipStream_t stream)
{
    const float* ice_h = (const float*)d_in[0];
    const float* p_w   = (const float*)d_in[1];
    const float* melt  = (const float*)d_in[2];
    const float* vel   = (const float*)d_in[3];
    const float* area  = (const float*)d_in[4];
    const float* grad  = (const float*)d_in[5];
    const float* flux  = (const float*)d_in[6];
    const int*   head  = (const int*)d_in[7];
    const int*   tail  = (const int*)d_in[8];
    float*       out   = (float*)d_out;

    const int n = in_sizes[0];   // nodes
    const int L = in_sizes[3];   // links

    fx4*   rec  = (fx4*)d_ws;
    float* recf = (float*)d_ws;

    const int BLK = 256;  // 8 wave32 waves per block

    // --- node prep ---
    int nv = n / 4, nrem = n - nv * 4;
    if (nv > 0)
        prep_v4<<<dim3((nv + BLK - 1) / BLK), BLK, 0, stream>>>(
            (const fx4*)ice_h, (const fx4*)p_w, rec, nv);
    if (nrem > 0)
        prep_tail<<<dim3(1), 64, 0, stream>>>(ice_h, p_w, rec, nv * 4, n);

    // --- link pass ---
    int lv = L / 4, lrem = L - lv * 4;
    if (lv > 0)
        link_v4<<<dim3((lv + BLK - 1) / BLK), BLK, 0, stream>>>(
            (const fx4*)vel, (const fx4*)area, (const fx4*)grad,
            (const fx4*)flux, (const ix4*)head, (const ix4*)tail,
            recf, lv);
    if (lrem > 0)
        link_tail<<<dim3(1), 64, 0, stream>>>(vel, area, grad, flux,
                                              head, tail, recf, lv * 4, L);

    // --- finalize ---
    if (nv > 0)
        finalize_v4<<<dim3((nv + BLK - 1) / BLK), BLK, 0, stream>>>(
            rec, (const fx4*)melt, (fx4*)out, nv);
    if (nrem > 0)
        finalize_tail<<<dim3(1), 64, 0, stream>>>(rec, melt, out, nv * 4, n);
}